// BoTorchGPWorldModel_27616639713803
// MI455X (gfx1250) — compile-verified
//
#include <hip/hip_runtime.h>
#include <hip/hip_bf16.h>

#define OBS 8
#define ACT 4
#define DIN 12
#define NTRAIN 128
#define BATCH 8
#define KDIM 24
#define LDA 25   // LDS row stride (odd => conflict-free for 16-lane column reads)

typedef __attribute__((ext_vector_type(2))) float v2f;
typedef __attribute__((ext_vector_type(8))) float v8f;

// workspace layout (float offsets)
#define WS_JM   0
#define WS_JV   (WS_JM + BATCH*DIN)                 // B*144
#define WS_PM   (WS_JV + BATCH*DIN*DIN)             // B*8
#define WS_CC   (WS_PM + BATCH*OBS)                 // B*12*8, cc[b][d][e]
#define WS_MAIN (WS_CC + BATCH*DIN*OBS)             // [a][b][batch] 8*8*8
#define WS_TR   (WS_MAIN + OBS*OBS*BATCH)           // [e][batch] 8*8

// ---------------- kernel 0: joint mean / joint covariance -------------------
__global__ __launch_bounds__(144) void k0_joint(const float* __restrict__ m_x,
                                                const float* __restrict__ s_x,
                                                const float* __restrict__ m_u,
                                                const float* __restrict__ s_u,
                                                const float* __restrict__ c_xu,
                                                float* __restrict__ ws) {
  int bt = blockIdx.x;
  int t = threadIdx.x;                 // 0..143
  if (t < DIN) {
    ws[WS_JM + bt*DIN + t] = (t < OBS) ? m_x[bt*OBS + t] : m_u[bt*ACT + (t-OBS)];
  }
  int r = t / DIN, c = t % DIN;
  float v;
  if (r < OBS && c < OBS) {
    v = s_x[bt*64 + r*8 + c];
  } else if (r < OBS) {                // upper-right: s_x @ c_xu
    float acc = 0.f;
    for (int k = 0; k < OBS; ++k) acc += s_x[bt*64 + r*8 + k] * c_xu[bt*32 + k*ACT + (c-OBS)];
    v = acc;
  } else if (c < OBS) {                // lower-left: (s_x @ c_xu)^T
    float acc = 0.f;
    for (int k = 0; k < OBS; ++k) acc += s_x[bt*64 + c*8 + k] * c_xu[bt*32 + k*ACT + (r-OBS)];
    v = acc;
  } else {
    v = s_u[bt*16 + (r-OBS)*ACT + (c-OBS)];
  }
  ws[WS_JV + bt*144 + r*DIN + c] = v;
}

// ---------------- kernel 1: predictive mean + cross covariance --------------
__global__ __launch_bounds__(128) void k1_pred(const float* __restrict__ X_train,
                                               const float* __restrict__ ls,
                                               const float* __restrict__ variances,
                                               const float* __restrict__ beta,
                                               float* __restrict__ ws) {
  int bt = blockIdx.x / OBS;
  int e  = blockIdx.x % OBS;
  int tid = threadIdx.x;
  __shared__ float jv[144], jm[12], ils[12];
  __shared__ float Mw[144], Binv[144];
  __shared__ float detS, red[13];

  for (int i = tid; i < 144; i += 128) jv[i] = ws[WS_JV + bt*144 + i];
  if (tid < 12) { jm[tid] = ws[WS_JM + bt*12 + tid]; ils[tid] = 1.0f / ls[e*12 + tid]; }
  if (tid < 13) red[tid] = 0.f;
  if (tid == 0) detS = 1.f;
  __syncthreads();
  for (int i = tid; i < 144; i += 128) {
    int r = i / 12, c = i % 12;
    Mw[i]   = ils[r] * jv[i] * ils[c] + ((r == c) ? 1.f : 0.f);
    Binv[i] = (r == c) ? 1.f : 0.f;
  }
  __syncthreads();
  // Gauss-Jordan (SPD, no pivoting); track det
  for (int k = 0; k < 12; ++k) {
    if (tid == 0) {
      float piv = Mw[k*12 + k]; detS *= piv; float ip = 1.f / piv;
      for (int c = 0; c < 12; ++c) { Mw[k*12 + c] *= ip; Binv[k*12 + c] *= ip; }
    }
    __syncthreads();
    if (tid < 12 && tid != k) {
      float f = Mw[tid*12 + k];
      for (int c = 0; c < 12; ++c) { Mw[tid*12 + c] -= f*Mw[k*12 + c]; Binv[tid*12 + c] -= f*Binv[k*12 + c]; }
    }
    __syncthreads();
  }
  // one training point per thread
  int n = tid;
  float invN[12], tv[12];
  for (int d = 0; d < 12; ++d) invN[d] = (X_train[n*12 + d] - jm[d]) * ils[d];
  float s = 0.f;
  for (int d = 0; d < 12; ++d) {
    float acc = 0.f;
    for (int c = 0; c < 12; ++c) acc += Binv[d*12 + c] * invN[c];
    tv[d] = acc; s += invN[d] * acc;
  }
  float lbn = __expf(-0.5f * s) * beta[e*NTRAIN + n];
  atomicAdd(&red[0], lbn);
  for (int d = 0; d < 12; ++d) atomicAdd(&red[1 + d], tv[d] * ils[d] * lbn);
  __syncthreads();
  if (tid == 0) {
    float cb = variances[e] / sqrtf(detS);
    ws[WS_PM + bt*8 + e] = cb * red[0];
    for (int d = 0; d < 12; ++d) ws[WS_CC + bt*96 + d*8 + e] = cb * red[1 + d];
  }
}

// ---------------- kernel 2: pair kernel (WMMA f32 16x16x4) ------------------
__global__ __launch_bounds__(256) void k2_pair(const float* __restrict__ X_train,
                                               const float* __restrict__ ls,
                                               const float* __restrict__ variances,
                                               const float* __restrict__ beta,
                                               const float* __restrict__ invK,
                                               float* __restrict__ ws) {
  int blk = blockIdx.x;
  int p  = blk >> 6;          // outer vmap (a)
  int q  = (blk >> 3) & 7;    // inner vmap (b)
  int bt = blk & 7;           // batch
  int tid  = threadIdx.x;
  int lane = tid & 31, wave = tid >> 5;
  int half = lane >> 4, l15 = lane & 15;

  __shared__ float jv[144], jm[12];
  __shared__ float la[12], lbv[12], lam[12], ild[12];
  __shared__ float Mw[144], S[144];
  __shared__ float A_ls[128*LDA];
  __shared__ float B_ls[128*LDA];
  __shared__ float ru[128], rv[128], bp[128], bq[128];   // ru/rv pre-halved
  __shared__ float detS, cabS, redM, redT;

  for (int i = tid; i < 144; i += 256) jv[i] = ws[WS_JV + bt*144 + i];
  if (tid < 12) {
    jm[tid] = ws[WS_JM + bt*12 + tid];
    float lsa = ls[p*12 + tid], lsb = ls[q*12 + tid];
    float l2a = lsa*lsa, l2b = lsb*lsb;
    float ia = 1.f/l2a, ib = 1.f/l2b;
    la[tid] = ia; lbv[tid] = ib;
    lam[tid] = 1.f / (ia + ib);       // Lambda
    ild[tid] = 1.f / (l2a + l2b);     // for exp1
  }
  if (tid < 128) { bp[tid] = beta[p*128 + tid]; bq[tid] = beta[q*128 + tid]; }
  if (tid == 0) { redM = 0.f; redT = 0.f; detS = 1.f; }
  __syncthreads();
  for (int i = tid; i < 144; i += 256) {
    int r = i / 12, c = i % 12;
    Mw[i] = jv[i] + ((r == c) ? lam[r] : 0.f);   // M = joint_var + diag(Lam)
    S[i]  = (r == c) ? 1.f : 0.f;
  }
  __syncthreads();
  for (int k = 0; k < 12; ++k) {
    if (tid == 0) {
      float piv = Mw[k*12 + k]; detS *= piv; float ip = 1.f / piv;
      for (int c = 0; c < 12; ++c) { Mw[k*12 + c] *= ip; S[k*12 + c] *= ip; }
    }
    __syncthreads();
    if (tid < 12 && tid != k) {
      float f = Mw[tid*12 + k];
      for (int c = 0; c < 12; ++c) { Mw[tid*12 + c] -= f*Mw[k*12 + c]; S[tid*12 + c] -= f*S[k*12 + c]; }
    }
    __syncthreads();
  }
  if (tid == 0) {
    // det(R) = det(M) * prod(inv_sum); c_ab = var_a*var_b/sqrt(det R)
    float pr = 1.f;
    for (int d = 0; d < 12; ++d) pr *= (la[d] + lbv[d]);
    cabS = variances[p] * variances[q] * rsqrtf(detS * pr);
  }
  // build A/B panels: A row i = [S u_i, -X_i*ild], B row j = [v_j, X_j]
  {
    int i = tid & 127;
    bool uside = (tid < 128);
    float x[12], vec[12];
    for (int d = 0; d < 12; ++d) x[d] = X_train[i*12 + d];
    float ei = 0.f;
    for (int d = 0; d < 12; ++d) ei += x[d]*x[d]*ild[d];
    if (uside) for (int d = 0; d < 12; ++d) vec[d] = lam[d]*la[d]*x[d]  - jm[d];
    else       for (int d = 0; d < 12; ++d) vec[d] = lam[d]*lbv[d]*x[d];
    float w[12], qf = 0.f;
    for (int d = 0; d < 12; ++d) {
      float acc = 0.f;
      for (int c = 0; c < 12; ++c) acc += S[d*12 + c] * vec[c];
      w[d] = acc; qf += vec[d] * acc;
    }
    if (uside) {
      for (int d = 0; d < 12; ++d) { A_ls[i*LDA + d] = w[d]; A_ls[i*LDA + 12 + d] = -x[d]*ild[d]; }
      ru[i] = 0.5f * (qf + ei);
    } else {
      for (int d = 0; d < 12; ++d) { B_ls[i*LDA + d] = vec[d]; B_ls[i*LDA + 12 + d] = x[d]; }
      rv[i] = 0.5f * (qf + ei);
    }
  }
  __syncthreads();

  // 128x128 GEMM (K=24) via V_WMMA_F32_16X16X4_F32: wave w owns row-tile w.
  const float cab = cabS;
  const int mrow  = wave * 16;
  const int ibase = mrow + 8*half;     // i index for r=0
  float mainAcc = 0.f, trAcc = 0.f;

  auto tile_gemm = [&](int ntile) -> v8f {
    v8f acc = {};
#pragma unroll
    for (int kk = 0; kk < 6; ++kk) {
      int k0 = kk*4 + 2*half;
      v2f a, b;
      a.x = A_ls[(mrow + l15)*LDA + k0];
      a.y = A_ls[(mrow + l15)*LDA + k0 + 1];
      b.x = B_ls[(ntile*16 + l15)*LDA + k0];
      b.y = B_ls[(ntile*16 + l15)*LDA + k0 + 1];
      acc = __builtin_amdgcn_wmma_f32_16x16x4_f32(false, a, false, b, (short)0, acc, false, false);
    }
    return acc;
  };

  if (p == q) {
    // diagonal pair: also accumulate trace(inv_K[p] @ Q^T)
    const float* kpbase = invK + (size_t)p*16384 + ibase;  // 32B-aligned
    for (int ntile = 0; ntile < 8; ++ntile) {
      v8f acc = tile_gemm(ntile);
      int j = ntile*16 + l15;
      float rvj = rv[j], bqj = bq[j];
      const float4* kp = (const float4*)(kpbase + (size_t)j*128);
      float4 kv0 = kp[0], kv1 = kp[1];     // invK[p][j][ibase .. ibase+7]
      float kvals[8] = {kv0.x, kv0.y, kv0.z, kv0.w, kv1.x, kv1.y, kv1.z, kv1.w};
#pragma unroll
      for (int r = 0; r < 8; ++r) {
        float qv = cab * __expf(-(ru[ibase + r] + rvj + acc[r]));
        mainAcc += bp[ibase + r] * qv * bqj;
        trAcc   += kvals[r] * qv;
      }
    }
  } else {
    for (int ntile = 0; ntile < 8; ++ntile) {
      v8f acc = tile_gemm(ntile);
      int j = ntile*16 + l15;
      float rvj = rv[j], bqj = bq[j];
#pragma unroll
      for (int r = 0; r < 8; ++r) {
        float qv = cab * __expf(-(ru[ibase + r] + rvj + acc[r]));
        mainAcc += bp[ibase + r] * qv * bqj;
      }
    }
  }

  // wave32 tree reduction, then one LDS atomic per wave
#pragma unroll
  for (int off = 16; off > 0; off >>= 1) {
    mainAcc += __shfl_xor(mainAcc, off, 32);
    trAcc   += __shfl_xor(trAcc,   off, 32);
  }
  if (lane == 0) {
    atomicAdd(&redM, mainAcc);
    if (p == q) atomicAdd(&redT, trAcc);
  }
  __syncthreads();
  if (tid == 0) {
    ws[WS_MAIN + (p*8 + q)*8 + bt] = redM;
    if (p == q) ws[WS_TR + p*8 + bt] = redT;
  }
}

// ---------------- kernel 3: assemble outputs --------------------------------
__global__ __launch_bounds__(64) void k3_final(const float* __restrict__ m_x,
                                               const float* __restrict__ s_x,
                                               const float* __restrict__ variances,
                                               const float* __restrict__ noises,
                                               const float* __restrict__ ws,
                                               float* __restrict__ out) {
  int bt = blockIdx.x;
  int t = threadIdx.x;
  int e = t >> 3, f = t & 7;
  const float* jvb = ws + WS_JV + bt*144;
  const float* ccb = ws + WS_CC + bt*96;
  float pm_e = ws[WS_PM + bt*8 + e];
  float pm_f = ws[WS_PM + bt*8 + f];
  float mef = ws[WS_MAIN + (e*8 + f)*8 + bt];
  float mfe = ws[WS_MAIN + (f*8 + e)*8 + bt];
  float pc = 0.5f*(mef + mfe) - pm_e*pm_f;
  if (e == f) pc += variances[e] - ws[WS_TR + e*8 + bt] + noises[e];
  float cxf = 0.f, cxr = 0.f;
  for (int d = 0; d < 12; ++d) {
    cxf += jvb[e*12 + d] * ccb[d*8 + f];   // cov_xf[e][f]
    cxr += jvb[f*12 + d] * ccb[d*8 + e];   // cov_xf[f][e]
  }
  float sxv = 0.5f*(s_x[bt*64 + e*8 + f] + s_x[bt*64 + f*8 + e]);
  float sv = sxv + pc + cxf + cxr;
  if (e == f) sv += 1e-8f;
  out[64 + bt*64 + e*8 + f] = sv;          // s_out
  if (f == 0) out[bt*8 + e] = m_x[bt*8 + e] + pm_e;  // m_out
}

extern "C" void kernel_launch(void* const* d_in, const int* in_sizes, int n_in,
                              void* d_out, int out_size, void* d_ws, size_t ws_size,
                              hipStream_t stream) {
  const float* m_x       = (const float*)d_in[0];
  const float* s_x       = (const float*)d_in[1];
  const float* m_u       = (const float*)d_in[2];
  const float* s_u       = (const float*)d_in[3];
  const float* c_xu      = (const float*)d_in[4];
  const float* X_train   = (const float*)d_in[5];
  const float* ls        = (const float*)d_in[6];
  const float* variances = (const float*)d_in[7];
  const float* noises    = (const float*)d_in[8];
  const float* invK      = (const float*)d_in[9];
  const float* beta      = (const float*)d_in[10];
  float* ws  = (float*)d_ws;
  float* out = (float*)d_out;

  k0_joint<<<BATCH, 144, 0, stream>>>(m_x, s_x, m_u, s_u, c_xu, ws);
  k1_pred<<<BATCH*OBS, 128, 0, stream>>>(X_train, ls, variances, beta, ws);
  k2_pair<<<OBS*OBS*BATCH, 256, 0, stream>>>(X_train, ls, variances, beta, invK, ws);
  k3_final<<<BATCH, 64, 0, stream>>>(m_x, s_x, variances, noises, ws, out);
}